// Giou_loss_52218212385067
// MI455X (gfx1250) — compile-verified
//
#include <hip/hip_runtime.h>

// CDNA5 / gfx1250: wave32, WMMA 16x16x4 f32 (exact fp32 matrix op) used for the
// deterministic final reduction.

typedef __attribute__((ext_vector_type(2))) float v2f;
typedef __attribute__((ext_vector_type(8))) float v8f;

constexpr int BATCH    = 128;
constexpr int WIDTH    = 256;
constexpr int HEIGHT   = 256;
constexpr int MAX_OBJS = 128;
constexpr int NITEMS   = BATCH * MAX_OBJS;   // 16384 boxes
constexpr int BLOCK    = 256;                 // 8 waves
constexpr int GRID     = NITEMS / BLOCK;      // 64 blocks -> 64 partial pairs

__device__ __forceinline__ float div_no_nan(float n, float d) {
    return (d != 0.0f) ? (n / d) : 0.0f;
}

// Stage 1: per-box GIoU * mask, deterministic LDS tree reduction per block.
__global__ void giou_partial_kernel(const float* __restrict__ loc_pred,
                                    const float* __restrict__ loc_true,
                                    const float* __restrict__ reg_mask,
                                    const int*   __restrict__ indices,
                                    float*       __restrict__ ws) {
    __shared__ float sl[BLOCK];
    __shared__ float sm[BLOCK];

    const int tid = threadIdx.x;
    const int i   = blockIdx.x * BLOCK + tid;   // box index in [0, 16384)

    // indices[i] = {bidx, pos, 0}
    const int bidx = indices[i * 3 + 0];
    const int pos  = indices[i * 3 + 1];

    // Gather pred box (16B-aligned float4) and fold in the /scale.
    // scale = [W,H,W,H] = 256 (power of two -> reciprocal multiply is exact).
    const float4 p = ((const float4*)loc_pred)[(long)bidx * (WIDTH * HEIGHT) + pos];
    const float4 t = ((const float4*)loc_true)[i];
    const float  m = reg_mask[i];

    const float invW = 1.0f / (float)WIDTH;
    const float invH = 1.0f / (float)HEIGHT;

    const float b1_ymin = p.x * invW, b1_xmin = p.y * invH;
    const float b1_ymax = p.z * invW, b1_xmax = p.w * invH;
    const float b2_ymin = t.x, b2_xmin = t.y;
    const float b2_ymax = t.z, b2_xmax = t.w;

    const float a1 = fmaxf(b1_ymax - b1_ymin, 0.0f) * fmaxf(b1_xmax - b1_xmin, 0.0f);
    const float a2 = fmaxf(b2_ymax - b2_ymin, 0.0f) * fmaxf(b2_xmax - b2_xmin, 0.0f);

    const float iy0 = fmaxf(b1_ymin, b2_ymin);
    const float ix0 = fmaxf(b1_xmin, b2_xmin);
    const float iy1 = fminf(b1_ymax, b2_ymax);
    const float ix1 = fminf(b1_xmax, b2_xmax);
    const float inter = fmaxf(iy1 - iy0, 0.0f) * fmaxf(ix1 - ix0, 0.0f);

    const float uni = a1 + a2 - inter;
    const float iou = div_no_nan(inter, uni);

    const float ey0 = fminf(b1_ymin, b2_ymin);
    const float ex0 = fminf(b1_xmin, b2_xmin);
    const float ey1 = fmaxf(b1_ymax, b2_ymax);
    const float ex1 = fmaxf(b1_xmax, b2_xmax);
    const float enc = fmaxf(ey1 - ey0, 0.0f) * fmaxf(ex1 - ex0, 0.0f);

    const float giou = iou - div_no_nan(enc - uni, enc);
    const float loss = 1.0f - giou;

    sl[tid] = loss * m;
    sm[tid] = m;
    __syncthreads();

    // Fixed-order tree reduction -> deterministic.
    #pragma unroll
    for (int s = BLOCK / 2; s > 0; s >>= 1) {
        if (tid < s) {
            sl[tid] += sl[tid + s];
            sm[tid] += sm[tid + s];
        }
        __syncthreads();
    }

    if (tid == 0) {
        ws[2 * blockIdx.x + 0] = sl[0];
        ws[2 * blockIdx.x + 1] = sm[0];
    }
}

// Stage 2: one wave32. Reduce the 64 (loss,mask) partials with two exact-f32
// V_WMMA_F32_16X16X4_F32 ops (A = partials in 16x4 layout, B = ones, C = 0).
// D[m][n] = rowsum_m of A; finishing: in-lane sum of the 8 D VGPRs covers rows
// held by this half-wave, lane0 + lane16 covers both halves. Total equals the
// sum of all 64 A entries independent of the exact (m,k) placement.
__global__ void giou_finalize_kernel(const float* __restrict__ ws,
                                     float* __restrict__ out) {
    const int lane = threadIdx.x;                 // 0..31, EXEC all ones
    // A 16x4 f32 layout: lanes 0-15 hold {K0,K1}, lanes 16-31 hold {K2,K3}.
    const int base = (lane < 16) ? lane : lane + 16;   // unique coverage of 0..63

    v2f aL, aM, bOne;
    aL[0] = ws[2 * base + 0];
    aL[1] = ws[2 * (base + 16) + 0];
    aM[0] = ws[2 * base + 1];
    aM[1] = ws[2 * (base + 16) + 1];
    bOne[0] = 1.0f;
    bOne[1] = 1.0f;

    v8f c = {};
    v8f dL = __builtin_amdgcn_wmma_f32_16x16x4_f32(
        /*neg_a=*/false, aL, /*neg_b=*/false, bOne,
        /*c_mod=*/(short)0, c, /*reuse_a=*/false, /*reuse_b=*/false);
    v8f dM = __builtin_amdgcn_wmma_f32_16x16x4_f32(
        /*neg_a=*/false, aM, /*neg_b=*/false, bOne,
        /*c_mod=*/(short)0, c, /*reuse_a=*/false, /*reuse_b=*/false);

    // Fixed-order in-lane sums of the 8 accumulator VGPRs.
    float tL = dL[0];
    float tM = dM[0];
    #pragma unroll
    for (int k = 1; k < 8; ++k) {
        tL += dL[k];
        tM += dM[k];
    }

    // Half-wave combine: lanes 0-15 hold rows 0-7, lanes 16-31 hold rows 8-15.
    const float totL = __shfl(tL, 0, 32) + __shfl(tL, 16, 32);
    const float totM = __shfl(tM, 0, 32) + __shfl(tM, 16, 32);

    if (lane == 0) {
        out[0] = totL / totM;   // sum(giou_loss * mask) / num_box
    }
}

extern "C" void kernel_launch(void* const* d_in, const int* in_sizes, int n_in,
                              void* d_out, int out_size, void* d_ws, size_t ws_size,
                              hipStream_t stream) {
    (void)in_sizes; (void)n_in; (void)out_size; (void)ws_size;

    const float* loc_pred = (const float*)d_in[0];
    const float* loc_true = (const float*)d_in[1];
    const float* reg_mask = (const float*)d_in[2];
    const int*   indices  = (const int*)d_in[3];
    float*       ws       = (float*)d_ws;
    float*       out      = (float*)d_out;

    giou_partial_kernel<<<GRID, BLOCK, 0, stream>>>(loc_pred, loc_true, reg_mask,
                                                    indices, ws);
    giou_finalize_kernel<<<1, 32, 0, stream>>>(ws, out);
}